// Conv_Embedding_2_add_39462159515869
// MI455X (gfx1250) — compile-verified
//
#include <hip/hip_runtime.h>
#include <hip/hip_bf16.h>

typedef __attribute__((ext_vector_type(2))) float v2f;
typedef __attribute__((ext_vector_type(8))) float v8f;

#define DDIM 512
#define KC 64            // K-chunk staged in LDS
#define BSTRIDE 160      // words per K-pair row; 160%64==32 -> halves on disjoint banks

// ---------------------------------------------------------------------------
// zero fill (float4 per thread)
// ---------------------------------------------------------------------------
__global__ __launch_bounds__(256) void zero_kernel(float* __restrict__ p, long long n4) {
    long long i = (long long)blockIdx.x * blockDim.x + threadIdx.x;
    if (i < n4) {
        *(float4*)(p + i * 4) = make_float4(0.f, 0.f, 0.f, 0.f);
    }
}

// ---------------------------------------------------------------------------
// C[M x 512] = A[M x 512] @ W[512 x 512] + bias via V_WMMA_F32_16X16X4_F32.
// Block = 256 threads (8 waves); block tile 128(M) x 64(N); wave owns one
// 16-row M strip and 4 N-tiles (A frag reused 4x per K-step).
//
// B staged in LDS, pair-interleaved: Bs[p*BSTRIDE + col*2 + {0,1}] =
// {W[2p][n0+col], W[2p+1][n0+col]}  ->  lane B-fragment = one ds_load_b64.
// Fragment layouts per ISA 7.12.2 (wave32):
//  A 16x4 f32: lane<16 -> K,K+1 ; lane>=16 -> K+2,K+3 (same M rows)
//  B 4x16 f32: VGPR0/1 = rows K/K+1 (lanes<16) and K+2/K+3 (lanes>=16)
//  C  16x16 f32: vgpr r -> M=r (+8 for lanes>=16), N=lane&15
// ---------------------------------------------------------------------------
__global__ __launch_bounds__(256) void gemm_bias_wmma(
    const float* __restrict__ A, const float* __restrict__ W,
    const float* __restrict__ bias, float* __restrict__ out, int M) {
    __shared__ float Bs[(KC / 2) * BSTRIDE];     // 32*160*4B = 20 KB

    const int lane = threadIdx.x & 31;
    const int wave = threadIdx.x >> 5;
    const int half = lane >> 4;      // 0/1
    const int l16  = lane & 15;
    const int m0 = blockIdx.x * 128 + wave * 16;
    const int n0 = blockIdx.y * 64;

    int arow = m0 + l16;
    if (arow >= M) arow = M - 1;                 // clamped read; store guarded
    const float* __restrict__ Arow = A + (size_t)arow * DDIM + half * 2;

    // staging: thread owns K-pair p and 8 contiguous cols
    const int p  = threadIdx.x >> 3;             // 0..31
    const int cg = (threadIdx.x & 7) * 8;        // 0..56
    float* const bdst = Bs + p * BSTRIDE + cg * 2;

    v8f acc[4] = {};
    for (int k0 = 0; k0 < DDIM; k0 += KC) {
        const float* w0 = W + (size_t)(k0 + 2 * p) * DDIM + n0 + cg;
        float4 r0a = *(const float4*)(w0);
        float4 r0b = *(const float4*)(w0 + 4);
        float4 r1a = *(const float4*)(w0 + DDIM);
        float4 r1b = *(const float4*)(w0 + DDIM + 4);
        __syncthreads();                          // prior chunk reads done
        ((v2f*)bdst)[0] = (v2f){r0a.x, r1a.x};
        ((v2f*)bdst)[1] = (v2f){r0a.y, r1a.y};
        ((v2f*)bdst)[2] = (v2f){r0a.z, r1a.z};
        ((v2f*)bdst)[3] = (v2f){r0a.w, r1a.w};
        ((v2f*)bdst)[4] = (v2f){r0b.x, r1b.x};
        ((v2f*)bdst)[5] = (v2f){r0b.y, r1b.y};
        ((v2f*)bdst)[6] = (v2f){r0b.z, r1b.z};
        ((v2f*)bdst)[7] = (v2f){r0b.w, r1b.w};
        __syncthreads();                          // chunk visible to all waves

#pragma unroll
        for (int kk = 0; kk < KC; kk += 4) {
            v2f a = *(const v2f*)(Arow + k0 + kk);
            const float* brow = Bs + ((kk >> 1) + half) * BSTRIDE + l16 * 2;
#pragma unroll
            for (int t = 0; t < 4; ++t) {
                v2f b = *(const v2f*)(brow + t * 32);
                acc[t] = __builtin_amdgcn_wmma_f32_16x16x4_f32(
                    false, a, false, b, (short)0, acc[t], false, false);
            }
        }
    }

    // epilogue: uniform branch keeps the common path straight-line
    if (blockIdx.x * 128 + 128 <= (unsigned)M) {
#pragma unroll
        for (int t = 0; t < 4; ++t) {
            int col = n0 + t * 16 + l16;
            float bv = bias[col];
#pragma unroll
            for (int r = 0; r < 8; ++r)
                out[(size_t)(m0 + r + half * 8) * DDIM + col] = acc[t][r] + bv;
        }
    } else {
#pragma unroll
        for (int t = 0; t < 4; ++t) {
            int col = n0 + t * 16 + l16;
            float bv = bias[col];
#pragma unroll
            for (int r = 0; r < 8; ++r) {
                int row = m0 + r + half * 8;
                if (row < M) out[(size_t)row * DDIM + col] = acc[t][r] + bv;
            }
        }
    }
}

// ---------------------------------------------------------------------------
// COO SpMM: acc[rows[e], :] += vals[e] * v[cols[e], :]
// one block (256 thr) per edge, 2 floats per thread, f32 global atomics
// ---------------------------------------------------------------------------
__global__ __launch_bounds__(256) void spmm_kernel(
    const float* __restrict__ v, const int* __restrict__ rows,
    const int* __restrict__ cols, const float* __restrict__ vals,
    float* __restrict__ acc, int E) {
    int e = blockIdx.x;
    if (e >= E) return;
    int d = threadIdx.x * 2;
    int c = cols[e];
    int r = rows[e];
    float w = vals[e];
    v2f s = *(const v2f*)(v + (size_t)c * DDIM + d);
    float* dst = acc + (size_t)r * DDIM + d;
    atomicAdd(dst + 0, w * s.x);
    atomicAdd(dst + 1, w * s.y);
}

// ---------------------------------------------------------------------------
// rowwise LN helpers: wave32 shuffle reduce + LDS cross-wave combine
// ---------------------------------------------------------------------------
__device__ __forceinline__ void block_reduce2(float& s, float& q, int t) {
    __shared__ float reds[4], redq[4];
#pragma unroll
    for (int off = 16; off > 0; off >>= 1) {
        s += __shfl_xor(s, off);
        q += __shfl_xor(q, off);
    }
    int w = t >> 5, l = t & 31;
    if (l == 0) { reds[w] = s; redq[w] = q; }
    __syncthreads();
    s = reds[0] + reds[1] + reds[2] + reds[3];
    q = redq[0] + redq[1] + redq[2] + redq[3];
}

__global__ __launch_bounds__(128) void ln_relu_kernel(
    const float* __restrict__ in, const float* __restrict__ gamma,
    const float* __restrict__ beta, float* __restrict__ out) {
    int row = blockIdx.x;
    int t = threadIdx.x;
    float4 v = *(const float4*)(in + (size_t)row * DDIM + t * 4);
    v.x = fmaxf(v.x, 0.f); v.y = fmaxf(v.y, 0.f);
    v.z = fmaxf(v.z, 0.f); v.w = fmaxf(v.w, 0.f);
    float s = v.x + v.y + v.z + v.w;
    float q = v.x * v.x + v.y * v.y + v.z * v.z + v.w * v.w;
    block_reduce2(s, q, t);
    float mu  = s * (1.f / DDIM);
    float var = q * (1.f / DDIM) - mu * mu;
    float inv = rsqrtf(var + 1e-5f);
    float4 g = *(const float4*)(gamma + t * 4);
    float4 b = *(const float4*)(beta + t * 4);
    float4 o;
    o.x = (v.x - mu) * inv * g.x + b.x;
    o.y = (v.y - mu) * inv * g.y + b.y;
    o.z = (v.z - mu) * inv * g.z + b.z;
    o.w = (v.w - mu) * inv * g.w + b.w;
    *(float4*)(out + (size_t)row * DDIM + t * 4) = o;
}

// ---------------------------------------------------------------------------
// final: out[b] = x[b] in [1,N] ? LN(relu(agg[x-1]))*g2+b2 + h1[x-1] : 0
// ---------------------------------------------------------------------------
__global__ __launch_bounds__(128) void gather_ln_res_kernel(
    const int* __restrict__ xidx, const float* __restrict__ agg,
    const float* __restrict__ h1, const float* __restrict__ gamma,
    const float* __restrict__ beta, float* __restrict__ out, int Nn) {
    int b = blockIdx.x;
    int t = threadIdx.x;
    float* dst = out + (size_t)b * DDIM + t * 4;
    int xv = xidx[b];                            // uniform per block
    if (xv < 1 || xv >= Nn + 1) {
        *(float4*)dst = make_float4(0.f, 0.f, 0.f, 0.f);
        return;
    }
    int row = xv - 1;
    float4 v = *(const float4*)(agg + (size_t)row * DDIM + t * 4);
    v.x = fmaxf(v.x, 0.f); v.y = fmaxf(v.y, 0.f);
    v.z = fmaxf(v.z, 0.f); v.w = fmaxf(v.w, 0.f);
    float s = v.x + v.y + v.z + v.w;
    float q = v.x * v.x + v.y * v.y + v.z * v.z + v.w * v.w;
    block_reduce2(s, q, t);
    float mu  = s * (1.f / DDIM);
    float var = q * (1.f / DDIM) - mu * mu;
    float inv = rsqrtf(var + 1e-5f);
    float4 g  = *(const float4*)(gamma + t * 4);
    float4 be = *(const float4*)(beta + t * 4);
    float4 h  = *(const float4*)(h1 + (size_t)row * DDIM + t * 4);
    float4 o;
    o.x = (v.x - mu) * inv * g.x + be.x + h.x;
    o.y = (v.y - mu) * inv * g.y + be.y + h.y;
    o.z = (v.z - mu) * inv * g.z + be.z + h.z;
    o.w = (v.w - mu) * inv * g.w + be.w + h.w;
    *(float4*)dst = o;
}

// ---------------------------------------------------------------------------
extern "C" void kernel_launch(void* const* d_in, const int* in_sizes, int n_in,
                              void* d_out, int out_size, void* d_ws, size_t ws_size,
                              hipStream_t stream) {
    const int*   x     = (const int*)d_in[0];
    const float* embed = (const float*)d_in[1];
    const float* W1    = (const float*)d_in[2];
    const float* b1    = (const float*)d_in[3];
    const float* W2    = (const float*)d_in[4];
    const float* b2    = (const float*)d_in[5];
    const float* g1    = (const float*)d_in[6];
    const float* be1   = (const float*)d_in[7];
    const float* g2    = (const float*)d_in[8];
    const float* be2   = (const float*)d_in[9];
    const int*   Gr    = (const int*)d_in[10];
    const int*   Gc    = (const int*)d_in[11];
    const float* Gv    = (const float*)d_in[12];

    const int Nn = in_sizes[1] / DDIM;   // 50000
    const int E  = in_sizes[10];         // 400000
    const int B  = in_sizes[0];          // 65536

    float* proj = (float*)d_ws;
    float* agg  = proj + (size_t)Nn * DDIM;
    float* h1   = agg  + (size_t)Nn * DDIM;
    float* out  = (float*)d_out;

    long long nd4 = (long long)Nn * DDIM / 4;
    dim3 zgrid((unsigned)((nd4 + 255) / 256));
    dim3 ggrid((Nn + 127) / 128, DDIM / 64);

    // ---- layer 1 ----
    zero_kernel<<<zgrid, 256, 0, stream>>>(agg, nd4);
    gemm_bias_wmma<<<ggrid, 256, 0, stream>>>(embed, W1, b1, proj, Nn);
    spmm_kernel<<<E, 256, 0, stream>>>(proj, Gr, Gc, Gv, agg, E);
    ln_relu_kernel<<<Nn, 128, 0, stream>>>(agg, g1, be1, h1);

    // ---- layer 2 ----
    zero_kernel<<<zgrid, 256, 0, stream>>>(agg, nd4);
    gemm_bias_wmma<<<ggrid, 256, 0, stream>>>(h1, W2, b2, proj, Nn);
    spmm_kernel<<<E, 256, 0, stream>>>(proj, Gr, Gc, Gv, agg, E);

    // ---- fused LN2 + residual + gather ----
    gather_ln_res_kernel<<<B, 128, 0, stream>>>(x, agg, h1, g2, be2, out, Nn);
}